// GlobalLandmarkAttention_82626580841038
// MI455X (gfx1250) — compile-verified
//
#include <hip/hip_runtime.h>

// ---------------------------------------------------------------------------
// Landmark attention for (B=4, L=8192, H=16, D=128), landmark interval 64
// (M = 128 landmarks). Full fp32 via V_WMMA_F32_16X16X4_F32 (CDNA5 / gfx1250,
// wave32). One 256-thread block (8 waves) handles 128 Q rows of one (b,h).
// ---------------------------------------------------------------------------

typedef __attribute__((ext_vector_type(2))) float v2f;
typedef __attribute__((ext_vector_type(4))) float v4f;
typedef __attribute__((ext_vector_type(8))) float v8f;

#define B_    4
#define L_    8192
#define H_    16
#define D_    128
#define M_    128     // number of landmarks
#define LINT  64      // landmark interval
#define ROWS  128     // Q rows per workgroup
#define NW    8       // waves per workgroup
#define STR   132     // LDS row stride in floats (16B-aligned rows, bank-spread)

__global__ __launch_bounds__(256)
void landmark_attn_kernel(const float* __restrict__ qg,
                          const float* __restrict__ kg,
                          const float* __restrict__ vg,
                          float* __restrict__ og)
{
    __shared__ float sAB[ROWS * STR];  // Q tile, later W (softmax weights)
    __shared__ float sKV[M_  * STR];   // Kl (row-major), later Vl^T (d-major)

    const int tid  = threadIdx.x;
    const int wave = tid >> 5;
    const int lane = tid & 31;
    const int hl   = lane >> 4;    // lane half: selects K pair {0,1} vs {2,3}
    const int ln   = lane & 15;    // row/col within fragment

    const int ltile = blockIdx.x;  // 0..63
    const int h     = blockIdx.y;  // 0..15
    const int b     = blockIdx.z;  // 0..3

    const long rowBase = (long)ltile * ROWS;
    const int  myRow   = wave * 16;            // this wave's row strip in tile

    // ---- Stage landmark-K (natural layout) and Q tile into LDS ------------
    // 128 rows x 128 floats each = 4096 float4 loads, 256 threads x 16 iters.
    for (int i = 0; i < 16; ++i) {
        int idx = i * 256 + tid;               // float4 index
        int m   = idx >> 5;                    // 32 float4 per row
        int d4  = (idx & 31) << 2;
        const v4f kv = *(const v4f*)(kg + (((long)b * L_ + (long)m * LINT) * H_ + h) * D_ + d4);
        *(v4f*)(&sKV[m * STR + d4]) = kv;      // byte addr (m*132+d4)*4, 16B aligned

        int r = m;                             // same decomposition for Q tile
        const v4f qv = *(const v4f*)(qg + (((long)b * L_ + rowBase + r) * H_ + h) * D_ + d4);
        *(v4f*)(&sAB[r * STR + d4]) = qv;
    }
    __syncthreads();

    // ---- Phase 1: S = Q * Kl^T  (16x128 per wave, fp32 WMMA) --------------
    v8f S[8];
    for (int ct = 0; ct < 8; ++ct) {           // landmark column tile
        v8f acc = {};
        #pragma unroll
        for (int kb = 0; kb < D_; kb += 4) {
            v2f a = *(const v2f*)&sAB[(myRow + ln) * STR + kb + 2 * hl];     // Q row
            v2f bb = *(const v2f*)&sKV[(ct * 16 + ln) * STR + kb + 2 * hl];  // Kl row (=col of Kl^T)
            acc = __builtin_amdgcn_wmma_f32_16x16x4_f32(
                false, a, false, bb, (short)0, acc, false, false);
        }
        S[ct] = acc;
    }

    // ---- Softmax over the 128 landmark columns (per row) ------------------
    const float rscale = 0.0883883476483184f;  // 1/sqrt(128)
    #pragma unroll
    for (int ct = 0; ct < 8; ++ct)
        #pragma unroll
        for (int vr = 0; vr < 8; ++vr)
            S[ct][vr] *= rscale;

    #pragma unroll
    for (int vr = 0; vr < 8; ++vr) {           // row = myRow + vr + 8*hl
        float mx = S[0][vr];
        #pragma unroll
        for (int ct = 1; ct < 8; ++ct) mx = fmaxf(mx, S[ct][vr]);
        // reduce across the 16 lanes of this half (xor<=8 stays in-half)
        #pragma unroll
        for (int off = 1; off < 16; off <<= 1)
            mx = fmaxf(mx, __shfl_xor(mx, off, 32));
        float sum = 0.0f;
        #pragma unroll
        for (int ct = 0; ct < 8; ++ct) {
            float p = __expf(S[ct][vr] - mx);
            S[ct][vr] = p;
            sum += p;
        }
        #pragma unroll
        for (int off = 1; off < 16; off <<= 1)
            sum += __shfl_xor(sum, off, 32);
        float inv = 1.0f / sum;
        #pragma unroll
        for (int ct = 0; ct < 8; ++ct) S[ct][vr] *= inv;
    }

    // ---- Swap LDS contents: W into sAB, Vl^T into sKV ---------------------
    __syncthreads();                           // everyone done reading Q / Kl

    #pragma unroll
    for (int ct = 0; ct < 8; ++ct)
        #pragma unroll
        for (int vr = 0; vr < 8; ++vr)
            sAB[(myRow + vr + 8 * hl) * STR + ct * 16 + ln] = S[ct][vr];

    for (int i = 0; i < 16; ++i) {             // Vl stored transposed: sKV[d*STR+m]
        int idx = i * 256 + tid;
        int m   = idx >> 5;
        int d4  = (idx & 31) << 2;
        const v4f vv = *(const v4f*)(vg + (((long)b * L_ + (long)m * LINT) * H_ + h) * D_ + d4);
        sKV[(d4 + 0) * STR + m] = vv[0];
        sKV[(d4 + 1) * STR + m] = vv[1];
        sKV[(d4 + 2) * STR + m] = vv[2];
        sKV[(d4 + 3) * STR + m] = vv[3];
    }
    __syncthreads();

    // ---- Phase 2: O = W * Vl  (reduction over landmarks) ------------------
    for (int ct = 0; ct < 8; ++ct) {           // output d-column tile
        v8f acc = {};
        #pragma unroll
        for (int kb = 0; kb < M_; kb += 4) {
            v2f a = *(const v2f*)&sAB[(myRow + ln) * STR + kb + 2 * hl];     // W row
            v2f bb = *(const v2f*)&sKV[(ct * 16 + ln) * STR + kb + 2 * hl];  // Vl^T row = d column
            acc = __builtin_amdgcn_wmma_f32_16x16x4_f32(
                false, a, false, bb, (short)0, acc, false, false);
        }
        // store O: row = rowBase + myRow + vr + 8*hl, col d = ct*16 + ln
        #pragma unroll
        for (int vr = 0; vr < 8; ++vr) {
            long gr = rowBase + myRow + vr + 8 * hl;
            og[(((long)b * L_ + gr) * H_ + h) * D_ + ct * 16 + ln] = acc[vr];
        }
    }
}

extern "C" void kernel_launch(void* const* d_in, const int* in_sizes, int n_in,
                              void* d_out, int out_size, void* d_ws, size_t ws_size,
                              hipStream_t stream) {
    (void)in_sizes; (void)n_in; (void)out_size; (void)d_ws; (void)ws_size;
    const float* q = (const float*)d_in[0];
    const float* k = (const float*)d_in[1];
    const float* v = (const float*)d_in[2];
    float* out = (float*)d_out;

    dim3 grid(L_ / ROWS, H_, B_);   // (64, 16, 4)
    dim3 block(256);
    landmark_attn_kernel<<<grid, block, 0, stream>>>(q, k, v, out);
}